// NodeModelGNN_62766652064045
// MI455X (gfx1250) — compile-verified
//
#include <hip/hip_runtime.h>

// Problem constants (match reference)
#define N_NODES  50000
#define N_EDGES  800000
#define IN_DIM   64
#define EDGE_DIM 32
#define CAT_DIM  96      // IN_DIM + EDGE_DIM
#define OUT_DIM  64

typedef __attribute__((ext_vector_type(2))) float v2f;  // A / B fragments for wmma f32 16x16x4
typedef __attribute__((ext_vector_type(8))) float v8f;  // C / D accumulator (16x16 f32 / 32 lanes)

// ---------------------------------------------------------------------------
// 1) init: deg = 1.0 (self-loop contribution), agg = 0
// ---------------------------------------------------------------------------
__global__ void k_init(float* __restrict__ deg, float* __restrict__ agg) {
    const int stride = gridDim.x * blockDim.x;
    int idx = blockIdx.x * blockDim.x + threadIdx.x;
    for (int i = idx; i < N_NODES; i += stride) deg[i] = 1.0f;
    const long long total = (long long)N_NODES * CAT_DIM;
    for (long long i = idx; i < total; i += stride) agg[i] = 0.0f;
}

// ---------------------------------------------------------------------------
// 2) degree histogram over source nodes (row)
// ---------------------------------------------------------------------------
__global__ void k_degree(const long long* __restrict__ row, float* __restrict__ deg) {
    int e = blockIdx.x * blockDim.x + threadIdx.x;
    if (e < N_EDGES) atomicAdd(&deg[(int)row[e]], 1.0f);
}

// ---------------------------------------------------------------------------
// 3) deg -> deg^{-1/2} in place (deg >= 1 always, no zero guard needed)
// ---------------------------------------------------------------------------
__global__ void k_rsqrt(float* __restrict__ deg) {
    int i = blockIdx.x * blockDim.x + threadIdx.x;
    if (i < N_NODES) deg[i] = rsqrtf(deg[i]);
}

// ---------------------------------------------------------------------------
// 4) scatter-aggregate: agg[col] += norm * cat(x[row], edge_attr)
//    One block (96 threads = 3 wave32) per edge; thread d handles feature d.
//    Contiguous reads of x / edge_attr; contiguous f32 atomics into agg.
// ---------------------------------------------------------------------------
__global__ void k_scatter(const float* __restrict__ x,
                          const float* __restrict__ ea,
                          const long long* __restrict__ row,
                          const long long* __restrict__ col,
                          const float* __restrict__ dis,   // deg^{-1/2}
                          float* __restrict__ agg) {
    const int e = blockIdx.x;
    const int d = threadIdx.x;          // 0..95
    const int r = (int)row[e];
    const int c = (int)col[e];
    const float norm = dis[r] * dis[c];
    const float v = (d < IN_DIM) ? x[(long long)r * IN_DIM + d]
                                 : ea[(long long)e * EDGE_DIM + (d - IN_DIM)];
    atomicAdd(&agg[(long long)c * CAT_DIM + d], norm * v);
}

// ---------------------------------------------------------------------------
// 5) dense GEMM: out[50000,64] = agg[50000,96] @ W[96,64] + bias
//    One wave per 16x16 output tile; K swept in 24 steps of
//    v_wmma_f32_16x16x4_f32 (native f32 matrix pipe, full precision).
//
//    ISA lane layouts (CDNA5 ISA):
//      A 16x4 f32 : lanes 0-15 -> M=lane, K={0,1} in v[0],v[1]
//                   lanes 16-31 -> M=lane-16, K={2,3}
//      B 4x16 f32 : lanes 0-15 -> N=lane, K={0,1}; lanes 16-31 -> N=lane-16, K={2,3}
//      C/D 16x16  : v[j] = C[M = j + 8*(lane>=16)][N = lane%16]
// ---------------------------------------------------------------------------
__global__ void k_gemm_wmma(const float* __restrict__ agg,
                            const float* __restrict__ W,
                            const float* __restrict__ bias,
                            float* __restrict__ out) {
    const int lane = threadIdx.x & 31;
    const int wave = (blockIdx.x * blockDim.x + threadIdx.x) >> 5;

    const int numMTiles = N_NODES / 16;   // 3125 (divides exactly)
    const int numNTiles = OUT_DIM / 16;   // 4
    if (wave >= numMTiles * numNTiles) return;   // wave-uniform: EXEC stays full

    const int tm = wave / numNTiles;
    const int tn = wave % numNTiles;
    const int m0 = tm * 16;
    const int n0 = tn * 16;

    const int half = lane >> 4;           // 0: lanes 0-15, 1: lanes 16-31
    const int l    = lane & 15;

    v8f c = {};
    const float* aRow = agg + (long long)(m0 + l) * CAT_DIM;

#pragma unroll
    for (int k = 0; k < CAT_DIM; k += 4) {
        const int ka = k + half * 2;      // this lane's two K indices
        v2f a, b;
        a.x = aRow[ka + 0];
        a.y = aRow[ka + 1];
        b.x = W[(ka + 0) * OUT_DIM + n0 + l];
        b.y = W[(ka + 1) * OUT_DIM + n0 + l];
        // 8 args: (neg_a, A, neg_b, B, c_mod, C, reuse_a, reuse_b)
        c = __builtin_amdgcn_wmma_f32_16x16x4_f32(
                false, a, false, b, (short)0, c, false, false);
    }

    const float bb = bias[n0 + l];
#pragma unroll
    for (int j = 0; j < 8; ++j) {
        out[(long long)(m0 + half * 8 + j) * OUT_DIM + n0 + l] = c[j] + bb;
    }
}

// ---------------------------------------------------------------------------
// launch
// ---------------------------------------------------------------------------
extern "C" void kernel_launch(void* const* d_in, const int* in_sizes, int n_in,
                              void* d_out, int out_size, void* d_ws, size_t ws_size,
                              hipStream_t stream) {
    const float*     x    = (const float*)d_in[0];          // [50000,64]
    const float*     ea   = (const float*)d_in[1];          // [800000,32]
    const float*     W    = (const float*)d_in[2];          // [96,64]
    const float*     bias = (const float*)d_in[3];          // [64]
    const long long* ei   = (const long long*)d_in[4];      // [2,800000] int64
    const long long* row  = ei;
    const long long* col  = ei + N_EDGES;

    // workspace layout: deg/dis [50000] (padded to 50176 for 256B alignment),
    // then agg [50000*96] f32  (total ~19.4 MB)
    float* deg = (float*)d_ws;
    float* agg = deg + 50176;

    k_init  <<<4096, 256, 0, stream>>>(deg, agg);
    k_degree<<<(N_EDGES + 255) / 256, 256, 0, stream>>>(row, deg);
    k_rsqrt <<<(N_NODES + 255) / 256, 256, 0, stream>>>(deg);
    k_scatter<<<N_EDGES, CAT_DIM, 0, stream>>>(x, ea, row, col, deg, agg);

    const int tiles = (N_NODES / 16) * (OUT_DIM / 16);      // 12500 waves
    const int blocks = (tiles * 32 + 255) / 256;            // 8 waves/block
    k_gemm_wmma<<<blocks, 256, 0, stream>>>(agg, W, bias, (float*)d_out);
}